// RoBERTaCRFTagger_91182155694805
// MI455X (gfx1250) — compile-verified
//
#include <hip/hip_runtime.h>

#define Bsz  128
#define Lseq 256
#define Hdim 1024
#define Ttag 9
#define TP   16   // tag dim padded to 16 for the WMMA N dimension / aligned rows

typedef __attribute__((ext_vector_type(2))) float v2f;
typedef __attribute__((ext_vector_type(8))) float v8f;

// ---------------------------------------------------------------------------
// Phase 1: emissions[B*L, TP] = hidden[B*L, H] @ W[T, H]^T + b
// One wave computes one 16(row) x 16(tag) f32 tile via V_WMMA_F32_16X16X4_F32,
// accumulating K=4 per issue over H=1024 (256 WMMAs). Memory-bound: streams
// `hidden` exactly once (134 MB ~ 5.8us at 23.3 TB/s); W is cache-resident.
// unroll 8 -> up to 16 global_load_b64 in flight per wave for latency hiding.
// ---------------------------------------------------------------------------
__global__ __launch_bounds__(256) void emis_kernel(const float* __restrict__ hidden,
                                                   const float* __restrict__ W,
                                                   const float* __restrict__ bias,
                                                   float* __restrict__ emis) {
  const int lane  = threadIdx.x & 31;
  const int wave  = threadIdx.x >> 5;
  const int tile  = blockIdx.x * 8 + wave;     // 2048 tiles of 16 rows
  const int row_base = tile * 16;
  const int m     = lane & 15;                 // A row within tile
  const int n     = lane & 15;                 // B/C/D column (tag)
  const int khalf = (lane >> 4) << 1;          // 0 (lanes 0-15) or 2 (lanes 16-31)
  const int wrow  = (n < Ttag) ? n : (Ttag - 1);  // clamp: cols 9..15 never read

  const float* ap = hidden + (size_t)(row_base + m) * Hdim + khalf;
  const float* bp = W      + (size_t)wrow          * Hdim + khalf;

  v8f c = {};
  #pragma unroll 8
  for (int k0 = 0; k0 < Hdim; k0 += 4) {
    v2f a  = *(const v2f*)(ap + k0);   // A: K = khalf, khalf+1
    v2f bv = *(const v2f*)(bp + k0);   // B: K = khalf, khalf+1 for column n
    c = __builtin_amdgcn_wmma_f32_16x16x4_f32(
        /*neg_a=*/false, a, /*neg_b=*/false, bv,
        /*c_mod=*/(short)0, c, /*reuse_a=*/false, /*reuse_b=*/false);
  }

  const float bn   = (n < Ttag) ? bias[n] : 0.0f;
  const int   mrow = row_base + ((lane >> 4) << 3);   // +8 for upper half-wave
  #pragma unroll
  for (int v = 0; v < 8; ++v) {
    emis[(size_t)(mrow + v) * TP + n] = c[v] + bn;    // row M = v + 8*(lane>=16)
  }
}

// ---------------------------------------------------------------------------
// Phase 2: CRF numerator + forward algorithm. One wave per batch element.
// Lanes 0..8 carry the 9 state scores; cross-state gather via lane broadcast.
// The transition column for this lane is hoisted to registers so the 256-step
// dependent chain touches no LDS.
// ---------------------------------------------------------------------------
__global__ __launch_bounds__(32) void crf_kernel(const float* __restrict__ emis,
                                                 const float* __restrict__ startT,
                                                 const float* __restrict__ endT,
                                                 const float* __restrict__ trans,
                                                 const int*   __restrict__ labels,
                                                 const int*   __restrict__ lengths,
                                                 float*       __restrict__ llh) {
  const int b    = blockIdx.x;
  const int lane = threadIdx.x;
  const int jj   = lane & 15;

  __shared__ float trans_s[Ttag * 16];   // [prev][cur], cur padded to 16
  for (int idx = lane; idx < Ttag * 16; idx += 32) {
    const int r = idx >> 4, cc = idx & 15;
    trans_s[idx] = (cc < Ttag) ? trans[r * Ttag + cc] : 0.0f;
  }
  __syncthreads();

  // Hoist this lane's incoming-transition column: tr[i] = trans[i][jj]
  float tr[Ttag];
  #pragma unroll
  for (int i = 0; i < Ttag; ++i) tr[i] = trans_s[i * 16 + jj];

  const int    len = lengths[b];
  const float* em  = emis + (size_t)b * Lseq * TP;
  const float  NEG = -1.0e30f;

  float score = (lane < Ttag) ? (startT[lane] + em[lane]) : NEG;

  for (int l = 1; l < Lseq; ++l) {
    float v[Ttag];
    float mx = NEG;
    #pragma unroll
    for (int i = 0; i < Ttag; ++i) {
      const float si = __shfl(score, i, 32);        // broadcast score[i]
      v[i] = si + tr[i];
      mx = fmaxf(mx, v[i]);
    }
    float s = 0.0f;
    #pragma unroll
    for (int i = 0; i < Ttag; ++i) s += __expf(v[i] - mx);
    const float nxt = mx + __logf(s) + em[l * TP + jj];
    score = (lane < Ttag && l < len) ? nxt : score; // masked update
  }

  // denominator: logsumexp_j(score_j + end_j)
  float t  = (lane < Ttag) ? (score + endT[lane]) : NEG;
  float mx = t;
  #pragma unroll
  for (int off = 16; off > 0; off >>= 1) mx = fmaxf(mx, __shfl_xor(mx, off, 32));
  float s = __expf(t - mx);                         // NEG lanes underflow to 0
  #pragma unroll
  for (int off = 16; off > 0; off >>= 1) s += __shfl_xor(s, off, 32);
  const float denom = mx + __logf(s);

  // numerator: gold-path score, lane-parallel over timesteps
  const int* lb = labels + b * Lseq;
  float part = 0.0f;
  for (int l = 1 + lane; l < Lseq; l += 32) {
    if (l < len) {
      const int prev = lb[l - 1], cur = lb[l];
      part += trans_s[prev * 16 + cur] + em[l * TP + cur];
    }
  }
  if (lane == 0) {
    const int t0 = lb[0];
    part += startT[t0] + em[t0];          // start + emission at step 0
    part += endT[lb[len - 1]];            // end transition at last valid step
  }
  #pragma unroll
  for (int off = 16; off > 0; off >>= 1) part += __shfl_xor(part, off, 32);

  if (lane == 0) llh[b] = part - denom;
}

// ---------------------------------------------------------------------------
// Phase 3: out = -mean(llh). Overwrites d_out (graph-replay safe).
// ---------------------------------------------------------------------------
__global__ __launch_bounds__(128) void reduce_kernel(const float* __restrict__ llh,
                                                     float* __restrict__ out) {
  __shared__ float s[128];
  const int t = threadIdx.x;
  s[t] = llh[t];
  __syncthreads();
  for (int off = 64; off > 0; off >>= 1) {
    if (t < off) s[t] += s[t + off];
    __syncthreads();
  }
  if (t == 0) out[0] = -s[0] / (float)Bsz;
}

extern "C" void kernel_launch(void* const* d_in, const int* in_sizes, int n_in,
                              void* d_out, int out_size, void* d_ws, size_t ws_size,
                              hipStream_t stream) {
  const float* hidden  = (const float*)d_in[0];
  const float* W       = (const float*)d_in[1];
  const float* bias    = (const float*)d_in[2];
  const float* startT  = (const float*)d_in[3];
  const float* endT    = (const float*)d_in[4];
  const float* trans   = (const float*)d_in[5];
  const int*   labels  = (const int*)d_in[6];
  const int*   lengths = (const int*)d_in[7];

  float* emis = (float*)d_ws;                              // 32768*16 f32 = 2 MB
  float* llh  = emis + (size_t)Bsz * Lseq * TP;            // 128 f32

  const int tiles = (Bsz * Lseq) / 16;                     // 2048
  emis_kernel<<<dim3(tiles / 8), dim3(256), 0, stream>>>(hidden, W, bias, emis);
  crf_kernel<<<dim3(Bsz), dim3(32), 0, stream>>>(emis, startT, endT, trans,
                                                 labels, lengths, llh);
  reduce_kernel<<<dim3(1), dim3(128), 0, stream>>>(llh, (float*)d_out);
}